// SegTransformerDecoder_48584670052364
// MI455X (gfx1250) — compile-verified
//
#include <hip/hip_runtime.h>

typedef __attribute__((ext_vector_type(2))) float v2f;
typedef __attribute__((ext_vector_type(8))) float v8f;

constexpr int kHb = 128, kWb = 128, kC = 128;
constexpr int kQ  = kHb * kWb;      // 16384
constexpr int kNC = 6;              // cameras
constexpr int kP  = 4;              // points
constexpr int kHf = 32, kWf = 88;   // feature map
constexpr float kEps = 1e-5f;

__device__ __forceinline__ v8f wmma4(v2f a, v2f b, v8f c) {
  // D = A(16x4 f32) * B(4x16 f32) + C(16x16 f32)
  return __builtin_amdgcn_wmma_f32_16x16x4_f32(false, a, false, b, (short)0, c,
                                               false, false);
}

// Async global->LDS copy of one DWORD per lane (CDNA5 GLOBAL_LOAD_ASYNC_TO_LDS).
// lds_off: per-lane LDS byte offset; gp: per-lane global address.
__device__ __forceinline__ void async_g2lds_b32(int lds_off, const float* gp) {
  asm volatile("global_load_async_to_lds_b32 %0, %1, off"
               :: "v"(lds_off), "v"(gp) : "memory");
}
__device__ __forceinline__ void async_wait0() {
  asm volatile("s_wait_asynccnt 0" ::: "memory");
}

// ---------------- tiled NCHW -> NHWC transpose (async LDS staging) ----------------
__global__ void k_transpose_in(const float* __restrict__ src, float* __restrict__ dst) {
  __shared__ float tile[32][33];          // only LDS object -> base offset 0
  int lane = threadIdx.x & 31;
  int row8 = threadIdx.x >> 5;            // 0..7
  int tq0 = blockIdx.x * 32;              // q tile
  int tc0 = blockIdx.y * 32;              // c tile
  #pragma unroll
  for (int j = 0; j < 4; ++j) {
    int r = row8 + j * 8;
    async_g2lds_b32((r * 33 + lane) * 4,
                    src + (size_t)(tc0 + r) * kQ + tq0 + lane);  // coalesced in q
  }
  async_wait0();
  __syncthreads();
  #pragma unroll
  for (int j = 0; j < 4; ++j) {
    int r = row8 + j * 8;
    dst[(size_t)(tq0 + r) * kC + tc0 + lane] = tile[lane][r];    // coalesced in c
  }
}

// ---------------- weight transpose: [tap][ci][co] -> [tap][co][ci] ----------------
__global__ void k_transpose_w(const float* __restrict__ w, float* __restrict__ wt) {
  size_t i = (size_t)blockIdx.x * 256 + threadIdx.x;   // 9*128*128
  int co  = (int)(i & 127);
  int ci  = (int)((i >> 7) & 127);
  int tap = (int)(i >> 14);
  wt[((size_t)tap * kC + co) * kC + ci] = w[i];
}

// ---------------- residual 3x3 conv via WMMA (weights pre-transposed) ----------------
__global__ void k_conv3x3_res(const float* __restrict__ qin,
                              const float* __restrict__ wt,   // [tap][co][ci]
                              const float* __restrict__ bias,
                              float* __restrict__ qout) {
  int gw   = blockIdx.x * 8 + (threadIdx.x >> 5);  // 8192 waves total
  int lane = threadIdx.x & 31;
  int hi   = lane >> 4;
  int l16  = lane & 15;
  int nt    = gw & 7;            // co tile
  int wtile = (gw >> 3) & 7;     // w tile
  int h     = gw >> 6;           // 0..127
  int co0 = nt * 16, w0 = wtile * 16;

  const float* bbase = wt + ((size_t)co0 + l16) * kC + 2 * hi;
  v8f acc = {};
  for (int ky = 0; ky < 3; ++ky) {
    int ih = h + ky - 1;
    if (ih < 0 || ih >= kHb) continue;            // wave-uniform skip
    for (int kx = 0; kx < 3; ++kx) {
      int iw = w0 + l16 + kx - 1;
      float am = (iw >= 0 && iw < kWb) ? 1.f : 0.f;  // border mask (no branch)
      int iwc = min(max(iw, 0), kWb - 1);
      const float* arow = qin + ((size_t)ih * kWb + iwc) * kC + 2 * hi;
      const float* brow = bbase + (size_t)(ky * 3 + kx) * kC * kC;
      #pragma unroll 8
      for (int k = 0; k < kC; k += 4) {
        v2f a = *(const v2f*)(arow + k);
        a = a * am;
        v2f b = *(const v2f*)(brow + k);
        acc = wmma4(a, b, acc);
      }
    }
  }
  int co = co0 + l16;
  float bn = bias[co];
  #pragma unroll
  for (int v = 0; v < 8; ++v) {
    int w = w0 + v + 8 * hi;
    size_t o = ((size_t)h * kWb + w) * kC + co;
    qout[o] = qin[o] + acc[v] + bn;
  }
}

// ---------------- instance-norm stats (per channel over Q) ----------------
__global__ void k_inorm_stats(const float* __restrict__ x, float* __restrict__ stats) {
  __shared__ float ssum[256], ssq[256];
  int c = blockIdx.x;
  float s = 0.f, s2 = 0.f;
  for (int q = threadIdx.x; q < kQ; q += 256) {
    float v = x[(size_t)q * kC + c];
    s += v; s2 += v * v;
  }
  ssum[threadIdx.x] = s; ssq[threadIdx.x] = s2;
  __syncthreads();
  for (int st = 128; st > 0; st >>= 1) {
    if ((int)threadIdx.x < st) {
      ssum[threadIdx.x] += ssum[threadIdx.x + st];
      ssq[threadIdx.x]  += ssq[threadIdx.x + st];
    }
    __syncthreads();
  }
  if (threadIdx.x == 0) {
    float m = ssum[0] / (float)kQ;
    float var = ssq[0] / (float)kQ - m * m;
    stats[c] = m;
    stats[kC + c] = rsqrtf(var + kEps);
  }
}

__global__ void k_inorm_apply(const float* __restrict__ xin,
                              const float* __restrict__ stats,
                              float* __restrict__ xout) {
  size_t i = (size_t)blockIdx.x * 256 + threadIdx.x;
  int c = (int)(i & (kC - 1));
  xout[i] = (xin[i] - stats[c]) * stats[kC + c];
}

// ---------------- offset head + projection ----------------
// grids[((q*P+p)*6+n)*3] = {gx, gy, valid}
__global__ void k_project(const float* __restrict__ q2,
                          const float* __restrict__ off_w,
                          const float* __restrict__ off_b,
                          const float* __restrict__ bev_pos,
                          const float* __restrict__ l2i,
                          float* __restrict__ grids) {
  int q = blockIdx.x * 256 + threadIdx.x;
  if (q >= kQ) return;
  const float* row = q2 + (size_t)q * kC;

  float off[12];
  #pragma unroll
  for (int o = 0; o < 12; ++o) {
    float acc = off_b[o];
    const float* wr = off_w + (size_t)o * kC;
    for (int c = 0; c < kC; ++c) acc += row[c] * wr[c];
    off[o] = 1.f / (1.f + expf(-acc));
  }

  float rx = bev_pos[(size_t)q * 3 + 0] * 102.4f - 51.2f;
  float ry = bev_pos[(size_t)q * 3 + 1] * 102.4f - 51.2f;
  float rz = bev_pos[(size_t)q * 3 + 2] * 8.0f - 5.0f;
  const float exy = 0.25f + kEps;
  const float ez  = 4.0f + kEps;

  for (int p = 0; p < kP; ++p) {
    float px = rx + off[p * 3 + 0] * exy * 2.f - exy;
    float py = ry + off[p * 3 + 1] * exy * 2.f - exy;
    float pz = rz + off[p * 3 + 2] * ez  * 2.f - ez;
    for (int n = 0; n < kNC; ++n) {
      const float* L = l2i + (size_t)n * 16;
      float c0 = L[0] * px + L[1] * py + L[2]  * pz + L[3];
      float c1 = L[4] * px + L[5] * py + L[6]  * pz + L[7];
      float c2 = L[8] * px + L[9] * py + L[10] * pz + L[11];
      float denom = fmaxf(c2, kEps);
      float u = c0 / denom * (1.f / 704.f);
      float v = c1 / denom * (1.f / 256.f);
      bool valid = (c2 > kEps) && (u > 0.f) && (u < 1.f) && (v > 0.f) && (v < 1.f);
      float* g = grids + (((size_t)q * kP + p) * kNC + n) * 3;
      g[0] = u * (float)kWf - 0.5f;
      g[1] = v * (float)kHf - 0.5f;
      g[2] = valid ? 1.f : 0.f;
    }
  }
}

// ---------------- bilinear sampling, sum over cameras ----------------
__global__ void k_sample(const float* __restrict__ feat,
                         const float* __restrict__ grids,
                         float* __restrict__ xout) {
  int tid = blockIdx.x * 256 + threadIdx.x;     // Q*4*32 threads
  int cg = tid & 31;
  int p  = (tid >> 5) & 3;
  int q  = tid >> 7;
  int c0 = cg * 4;

  float a0 = 0.f, a1 = 0.f, a2 = 0.f, a3 = 0.f;
  for (int n = 0; n < kNC; ++n) {
    const float* g = grids + (((size_t)q * kP + p) * kNC + n) * 3;
    float valid = g[2];
    if (valid == 0.f) continue;
    float gx = g[0], gy = g[1];
    float x0f = floorf(gx), y0f = floorf(gy);
    int x0 = (int)x0f, y0 = (int)y0f;
    float wx1 = gx - x0f, wy1 = gy - y0f;
    float wxs[2] = {1.f - wx1, wx1};
    float wys[2] = {1.f - wy1, wy1};
    int xs[2] = {x0, x0 + 1};
    int ys[2] = {y0, y0 + 1};
    const float* fb = feat + ((size_t)n * kC + c0) * (kHf * kWf);
    #pragma unroll
    for (int jy = 0; jy < 2; ++jy)
      #pragma unroll
      for (int jx = 0; jx < 2; ++jx) {
        int xi = xs[jx], yi = ys[jy];
        if (xi < 0 || xi > kWf - 1 || yi < 0 || yi > kHf - 1) continue;
        float wgt = wxs[jx] * wys[jy] * valid;
        size_t base = (size_t)yi * kWf + xi;
        a0 += wgt * fb[base];
        a1 += wgt * fb[base + 1 * (size_t)(kHf * kWf)];
        a2 += wgt * fb[base + 2 * (size_t)(kHf * kWf)];
        a3 += wgt * fb[base + 3 * (size_t)(kHf * kWf)];
      }
  }
  float4* o = (float4*)(xout + (size_t)q * 512 + p * kC + c0);
  *o = make_float4(a0, a1, a2, a3);
}

// ---------------- WMMA GEMM: out = act(A @ W^T + bias) (+res) ----------------
// Each wave computes a 32x64 output tile (2 M-tiles x 4 N-tiles): 6 b64 loads
// per 8 WMMAs. MODE 0: exact GELU; MODE 1: add residual. Needs M%32==0, N%64==0.
template <int MODE>
__global__ void k_gemm_wmma(const float* __restrict__ A, const float* __restrict__ W,
                            const float* __restrict__ bias, const float* __restrict__ res,
                            float* __restrict__ out, int M, int N, int K) {
  int wave = (blockIdx.x * 256 + threadIdx.x) >> 5;
  int lane = threadIdx.x & 31;
  int hi = lane >> 4, l16 = lane & 15;
  int ngroups = N >> 6;                 // groups of 4 n-tiles
  int mt = wave / ngroups;              // 32-row tile
  int n0 = (wave % ngroups) * 64;

  const float* ar0 = A + ((size_t)(mt * 32 + l16)) * K + 2 * hi;
  const float* ar1 = ar0 + (size_t)16 * K;
  const float* br0 = W + ((size_t)(n0 + l16)) * K + 2 * hi;
  const float* br1 = br0 + (size_t)16 * K;
  const float* br2 = br0 + (size_t)32 * K;
  const float* br3 = br0 + (size_t)48 * K;

  v8f accs[8];
  #pragma unroll
  for (int i = 0; i < 8; ++i) accs[i] = (v8f){};

  #pragma unroll 2
  for (int k = 0; k < K; k += 4) {
    v2f a0 = *(const v2f*)(ar0 + k);
    v2f a1 = *(const v2f*)(ar1 + k);
    v2f b0 = *(const v2f*)(br0 + k);
    v2f b1 = *(const v2f*)(br1 + k);
    v2f b2 = *(const v2f*)(br2 + k);
    v2f b3 = *(const v2f*)(br3 + k);
    accs[0] = wmma4(a0, b0, accs[0]);
    accs[1] = wmma4(a0, b1, accs[1]);
    accs[2] = wmma4(a0, b2, accs[2]);
    accs[3] = wmma4(a0, b3, accs[3]);
    accs[4] = wmma4(a1, b0, accs[4]);
    accs[5] = wmma4(a1, b1, accs[5]);
    accs[6] = wmma4(a1, b2, accs[6]);
    accs[7] = wmma4(a1, b3, accs[7]);
  }

  #pragma unroll
  for (int tm = 0; tm < 2; ++tm) {
    #pragma unroll
    for (int tn = 0; tn < 4; ++tn) {
      int n = n0 + tn * 16 + l16;
      float bn = bias[n];
      #pragma unroll
      for (int v = 0; v < 8; ++v) {
        int m = mt * 32 + tm * 16 + v + 8 * hi;
        size_t o = (size_t)m * N + n;
        float val = accs[tm * 4 + tn][v] + bn;
        if (MODE == 0) {
          val = 0.5f * val * (1.f + erff(val * 0.70710678118654752f));
        } else {
          val += res[o];
        }
        out[o] = val;
      }
    }
  }
}

// ---------------- final inorm + tiled NHWC -> NCHW (async LDS staging) -------
__global__ void k_inorm_out(const float* __restrict__ x,
                            const float* __restrict__ stats,
                            float* __restrict__ out) {
  __shared__ float tile[32][33];          // only LDS object -> base offset 0
  int lane = threadIdx.x & 31;
  int row8 = threadIdx.x >> 5;
  int tq0 = blockIdx.x * 32;
  int tc0 = blockIdx.y * 32;
  #pragma unroll
  for (int j = 0; j < 4; ++j) {
    int r = row8 + j * 8;
    async_g2lds_b32((r * 33 + lane) * 4,
                    x + (size_t)(tq0 + r) * kC + tc0 + lane);    // coalesced in c
  }
  async_wait0();
  __syncthreads();
  #pragma unroll
  for (int j = 0; j < 4; ++j) {
    int r = row8 + j * 8;
    int c = tc0 + r;                       // channel of this output row
    out[(size_t)c * kQ + tq0 + lane] =
        (tile[lane][r] - stats[c]) * stats[kC + c];              // coalesced in q
  }
}

extern "C" void kernel_launch(void* const* d_in, const int* in_sizes, int n_in,
                              void* d_out, int out_size, void* d_ws, size_t ws_size,
                              hipStream_t stream) {
  (void)in_sizes; (void)n_in; (void)out_size; (void)ws_size;
  const float* feats = (const float*)d_in[0];
  const float* l2i   = (const float*)d_in[1];
  const float* bevq  = (const float*)d_in[2];
  const float* bevp  = (const float*)d_in[3];
  const float* w_in  = (const float*)d_in[4];
  const float* b_in  = (const float*)d_in[5];
  const float* off_w = (const float*)d_in[6];
  const float* off_b = (const float*)d_in[7];
  const float* w1    = (const float*)d_in[8];
  const float* b1    = (const float*)d_in[9];
  const float* w2    = (const float*)d_in[10];
  const float* b2    = (const float*)d_in[11];
  const float* w3    = (const float*)d_in[12];
  const float* b3    = (const float*)d_in[13];
  const float* w_out = (const float*)d_in[14];
  const float* b_out = (const float*)d_in[15];
  float* out = (float*)d_out;

  char* ws = (char*)d_ws;
  const size_t MB = 1024u * 1024u;
  float* q0    = (float*)(ws + 0);          // 8 MB, dead after conv1
  float* q1    = (float*)(ws + 8 * MB);     // 8 MB, normalized in place -> q2
  float* grids = (float*)(ws + 0);          // 4.7 MB, reuses q0 slot
  float* xbuf  = (float*)(ws + 16 * MB);    // 32 MB (Q x 512)
  float* h1    = (float*)(ws + 48 * MB);    // 32 MB
  float* h2    = xbuf;                      // reuses x after GEMM1
  float* r     = h1;                        // 8 MB, reuses h1 after GEMM2
  float* r2    = (float*)(ws + 56 * MB);    // 8 MB
  float* wtA   = (float*)(ws + 80 * MB);    // 576 KB (outside h1 region)
  float* wtB   = (float*)(ws + 81 * MB);    // 576 KB
  float* stats = (float*)(ws + 82 * MB);    // 1 KB

  // 0. weight transposes for both convs
  k_transpose_w<<<(9 * kC * kC) / 256, 256, 0, stream>>>(w_in, wtA);
  k_transpose_w<<<(9 * kC * kC) / 256, 256, 0, stream>>>(w_out, wtB);
  // 1. NCHW -> NHWC (async LDS-tiled)
  k_transpose_in<<<dim3(kQ / 32, kC / 32), 256, 0, stream>>>(bevq, q0);
  // 2. q1 = q0 + conv3x3(q0) + b    (WMMA)
  k_conv3x3_res<<<1024, 256, 0, stream>>>(q0, wtA, b_in, q1);
  // 3-4. instance norm (in place -> q2 == q1)
  k_inorm_stats<<<kC, 256, 0, stream>>>(q1, stats);
  k_inorm_apply<<<(kQ * kC) / 256, 256, 0, stream>>>(q1, stats, q1);
  // 5. offsets + projection
  k_project<<<kQ / 256, 256, 0, stream>>>(q1, off_w, off_b, bevp, l2i, grids);
  // 6. bilinear sampling -> x (Q x 512)
  k_sample<<<(kQ * kP * 32) / 256, 256, 0, stream>>>(feats, grids, xbuf);
  // 7-9. MLP (WMMA, 32x64 tiles per wave)
  k_gemm_wmma<0><<<(kQ / 32) * (512 / 64) / 8, 256, 0, stream>>>(xbuf, w1, b1, nullptr, h1, kQ, 512, 512);
  k_gemm_wmma<0><<<(kQ / 32) * (512 / 64) / 8, 256, 0, stream>>>(h1, w2, b2, nullptr, h2, kQ, 512, 512);
  k_gemm_wmma<1><<<(kQ / 32) * (128 / 64) / 8, 256, 0, stream>>>(h2, w3, b3, q1, r, kQ, 128, 512);
  // 10-11. instance norm (in place -> q3 == r)
  k_inorm_stats<<<kC, 256, 0, stream>>>(r, stats);
  k_inorm_apply<<<(kQ * kC) / 256, 256, 0, stream>>>(r, stats, r);
  // 12. r2 = q3 + conv3x3(q3) + b   (WMMA)
  k_conv3x3_res<<<1024, 256, 0, stream>>>(r, wtB, b_out, r2);
  // 13-14. instance norm + async LDS-tiled NHWC -> NCHW output
  k_inorm_stats<<<kC, 256, 0, stream>>>(r2, stats);
  k_inorm_out<<<dim3(kQ / 32, kC / 32), 256, 0, stream>>>(r2, stats, out);
}